// HGN_Basket_70300024701220
// MI455X (gfx1250) — compile-verified
//
#include <hip/hip_runtime.h>
#include <hip/hip_bf16.h>
#include <stdint.h>

// ---------------------------------------------------------------------------
// LightGCN-style propagation for HGN_Basket on MI455X (gfx1250).
// Memory-bound sparse scatter/gather: fp32, ~20 GB of traffic over 10 SpMM
// passes. Optimization target is data movement: CDNA5 async global->LDS
// gathers (global_load_async_to_lds_b128, ASYNCcnt-thresholded double
// buffering) + hardware global_atomic_add_f32 scatter. WMMA is inapplicable
// (no dense GEMM; ~0.25 FLOP/byte).
// ---------------------------------------------------------------------------

#define DIMS 128            // embedding dim
#define TILE_EDGES 16       // edges per wave (two async half-tiles of 8)
#define HALF_TILE 8
#define WAVES_PER_BLK 8     // 256-thread blocks = 8 wave32s

__device__ __forceinline__ void atom_add_f32(float* p, float v) {
  // lowers to global_atomic_add_f32 (no CAS loop)
  unsafeAtomicAdd(p, v);
}

__device__ __forceinline__ int rdlane_i(int v, int l) {
  return __builtin_amdgcn_readlane(v, l);          // v_readlane_b32
}
__device__ __forceinline__ float rdlane_f(float v, int l) {
  return __int_as_float(__builtin_amdgcn_readlane(__float_as_int(v), l));
}

// -------------------- elementwise helpers --------------------

__global__ void k_zero4(float4* __restrict__ p, long n4) {
  long i = (long)blockIdx.x * blockDim.x + threadIdx.x;
  if (i < n4) p[i] = make_float4(0.f, 0.f, 0.f, 0.f);
}

// x_prev = acc = concat(a, b); x_next = 0
__global__ void k_init_x0(const float4* __restrict__ a,
                          const float4* __restrict__ b,
                          long nA4, long n4,
                          float4* __restrict__ xp,
                          float4* __restrict__ xn,
                          float4* __restrict__ acc) {
  long i = (long)blockIdx.x * blockDim.x + threadIdx.x;
  if (i >= n4) return;
  float4 v = (i < nA4) ? a[i] : b[i - nA4];
  xp[i]  = v;
  acc[i] = v;
  xn[i]  = make_float4(0.f, 0.f, 0.f, 0.f);
}

// acc = (acc + x) * scale ; optionally zero z (next pass's scatter target)
__global__ void k_add_scale(float4* __restrict__ acc,
                            const float4* __restrict__ x,
                            float4* __restrict__ z,
                            long n4, float scale) {
  long i = (long)blockIdx.x * blockDim.x + threadIdx.x;
  if (i >= n4) return;
  float4 A = acc[i];
  float4 X = x[i];
  A.x = (A.x + X.x) * scale;
  A.y = (A.y + X.y) * scale;
  A.z = (A.z + X.z) * scale;
  A.w = (A.w + X.w) * scale;
  acc[i] = A;
  if (z) z[i] = make_float4(0.f, 0.f, 0.f, 0.f);
}

// -------------------- degree / normalization --------------------

__global__ void k_deg(const int* __restrict__ src, float* __restrict__ deg, int nE) {
  int e = blockIdx.x * blockDim.x + threadIdx.x;
  if (e < nE) atom_add_f32(&deg[src[e]], 1.0f);
}

__global__ void k_dinv(float* __restrict__ d, int n) {
  int i = blockIdx.x * blockDim.x + threadIdx.x;
  if (i < n) {
    float v = d[i];
    d[i] = (v > 0.f) ? rsqrtf(v) : 0.f;
  }
}

// -------------------- SpMM: x_out[src] += dinv[src]*dinv[dst] * x_in[dst] ----
// One wave per 16 edges. Lanes 0..15 fetch edge metadata (broadcast later via
// v_readlane). Two half-tiles of 8 async B128 gathers are kept in flight:
// issue 16, s_wait_asynccnt 8 -> drain first 8 rows into f32 global atomics
// while the second 8 are still streaming into LDS, then s_wait_asynccnt 0.

__global__ void k_spmm(const int* __restrict__ src,
                       const int* __restrict__ dst,
                       const float* __restrict__ dinv,
                       const float* __restrict__ x_in,
                       float* __restrict__ x_out,
                       int nE) {
  __shared__ float4 stage[WAVES_PER_BLK * TILE_EDGES * 32]; // 64 KB / block

  const int lane   = threadIdx.x & 31;
  const int waveId = threadIdx.x >> 5;
  const long tileBase = ((long)blockIdx.x * WAVES_PER_BLK + waveId) * TILE_EDGES;
  if (tileBase >= nE) return;

  // lanes 0..15 hold metadata for edge tileBase+lane
  int   s = 0, d = 0;
  float wgt = 0.f;
  if (lane < TILE_EDGES) {
    long e = tileBase + lane;
    if (e >= nE) e = nE - 1;             // clamp (loads benign; atomics guarded)
    s   = src[e];
    d   = dst[e];
    wgt = dinv[s] * dinv[d];
  }

  float4* myStage = &stage[waveId * TILE_EDGES * 32];
  const uint32_t ldsBase = (uint32_t)(uintptr_t)(void*)myStage;

  // issue all 16 async gathers: row x_in[dt] (512B) -> LDS slot t, 16B/lane
#pragma unroll
  for (int t = 0; t < TILE_EDGES; ++t) {
    int dt = rdlane_i(d, t);
    uint64_t gaddr = (uint64_t)(uintptr_t)(x_in + (long)dt * DIMS) +
                     (uint64_t)(lane * 16);
    uint32_t laddr = ldsBase + (uint32_t)((t * 32 + lane) * 16);
    asm volatile("global_load_async_to_lds_b128 %0, %1, off"
                 :: "v"(laddr), "v"(gaddr)
                 : "memory");
  }

  // first half-tile landed (async loads complete in order)
  asm volatile("s_wait_asynccnt 8" ::: "memory");

#pragma unroll
  for (int t = 0; t < HALF_TILE; ++t) {
    if (tileBase + t >= nE) break;       // wave-uniform tail guard
    int   st = rdlane_i(s, t);
    float wt = rdlane_f(wgt, t);
    float4 v = myStage[t * 32 + lane];
    float* outp = x_out + (long)st * DIMS + lane * 4;
    atom_add_f32(outp + 0, v.x * wt);
    atom_add_f32(outp + 1, v.y * wt);
    atom_add_f32(outp + 2, v.z * wt);
    atom_add_f32(outp + 3, v.w * wt);
  }

  // remaining half-tile
  asm volatile("s_wait_asynccnt 0" ::: "memory");

#pragma unroll
  for (int t = HALF_TILE; t < TILE_EDGES; ++t) {
    if (tileBase + t >= nE) break;       // wave-uniform tail guard
    int   st = rdlane_i(s, t);
    float wt = rdlane_f(wgt, t);
    float4 v = myStage[t * 32 + lane];
    float* outp = x_out + (long)st * DIMS + lane * 4;
    atom_add_f32(outp + 0, v.x * wt);
    atom_add_f32(outp + 1, v.y * wt);
    atom_add_f32(outp + 2, v.z * wt);
    atom_add_f32(outp + 3, v.w * wt);
  }
}

// -------------------- row gather: out[r] = table[idx[r]] --------------------

__global__ void k_gather(const int* __restrict__ idx,
                         const float* __restrict__ table,
                         float* __restrict__ out, int rows) {
  int r    = (blockIdx.x * blockDim.x + threadIdx.x) >> 5;
  int lane = threadIdx.x & 31;
  if (r >= rows) return;
  const float4* s = (const float4*)(table + (long)idx[r] * DIMS);
  float4*       o = (float4*)(out + (long)r * DIMS);
  o[lane] = s[lane];
}

// ---------------------------------------------------------------------------

extern "C" void kernel_launch(void* const* d_in, const int* in_sizes, int n_in,
                              void* d_out, int out_size, void* d_ws, size_t ws_size,
                              hipStream_t stream) {
  const float* users_emb   = (const float*)d_in[0];
  const float* items_emb   = (const float*)d_in[1];
  const float* baskets_emb = (const float*)d_in[2];
  const int*   baskets     = (const int*)d_in[3];
  const int*   pos_items   = (const int*)d_in[4];
  const int*   neg_items   = (const int*)d_in[5];
  const int*   users       = (const int*)d_in[6];
  const int*   e_u2i       = (const int*)d_in[7];
  const int*   e_b2i       = (const int*)d_in[8];
  const int*   e_u2i_aug   = (const int*)d_in[9];
  const int*   e_b2i_aug   = (const int*)d_in[10];

  const int Un    = in_sizes[0] / DIMS;   // 100000
  const int In    = in_sizes[1] / DIMS;   // 100000
  const int Bn    = in_sizes[2] / DIMS;   // 100000
  const int batch = in_sizes[3];          // 4096
  const int N     = Un + In;              // 200000 (== Bn + In)
  const long ND   = (long)N * DIMS;
  const long n4   = ND / 4;

  float* out    = (float*)d_out;
  float* x_prev = (float*)d_ws;           // N*128 floats
  float* x_next = x_prev + ND;            // N*128 floats
  float* dinv   = x_next + ND;            // N floats

  struct GraphCfg {
    const int* e; long nE; const float* a; const float* b; int K; long outOff;
  };
  const GraphCfg gs[4] = {
    { e_u2i,     in_sizes[7]  / 2, users_emb,   items_emb, 2, 0L * ND }, // ef1
    { e_u2i_aug, in_sizes[9]  / 2, users_emb,   items_emb, 2, 1L * ND }, // ef2
    { e_b2i,     in_sizes[8]  / 2, baskets_emb, items_emb, 3, 2L * ND }, // bf1
    { e_b2i_aug, in_sizes[10] / 2, baskets_emb, items_emb, 3, 3L * ND }, // bf2
  };

  const int T   = 256;
  const int bE4 = (int)((n4 + T - 1) / T);

  for (int g = 0; g < 4; ++g) {
    const GraphCfg& G = gs[g];
    const int* src = G.e;
    const int* dst = G.e + G.nE;
    float*     acc = out + G.outOff;
    const int  nE  = (int)G.nE;

    // degree -> dinv (in place)
    k_zero4<<<(N / 4 + T - 1) / T, T, 0, stream>>>((float4*)dinv, N / 4);
    k_deg<<<(nE + T - 1) / T, T, 0, stream>>>(src, dinv, nE);
    k_dinv<<<(N + T - 1) / T, T, 0, stream>>>(dinv, N);

    // x_prev = acc = x0; x_next = 0
    k_init_x0<<<bE4, T, 0, stream>>>((const float4*)G.a, (const float4*)G.b,
                                     ((long)Un * DIMS) / 4, n4,
                                     (float4*)x_prev, (float4*)x_next,
                                     (float4*)acc);

    float* xp = x_prev;
    float* xn = x_next;
    const int edgesPerBlk = WAVES_PER_BLK * TILE_EDGES;  // 128
    const int spmmBlocks  = (nE + edgesPerBlk - 1) / edgesPerBlk;
    for (int k = 1; k <= G.K; ++k) {
      k_spmm<<<spmmBlocks, T, 0, stream>>>(src, dst, dinv, xp, xn, nE);
      const float scale = (k == G.K) ? 1.0f / (float)(G.K + 1) : 1.0f;
      float* zbuf = (k < G.K) ? xp : nullptr;   // zero next scatter target
      k_add_scale<<<bE4, T, 0, stream>>>((float4*)acc, (const float4*)xn,
                                         (float4*)zbuf, n4, scale);
      float* t = xp; xp = xn; xn = t;
    }
  }

  // final gathers (tables live in d_out / inputs, already computed in-order)
  const float* items1   = out + 0L * ND + (long)Un * DIMS; // ef1 item rows
  const float* baskets1 = out + 2L * ND;                    // bf1 basket rows
  const float* bitems1  = out + 2L * ND + (long)Bn * DIMS;  // bf1 item rows

  const int gB = (batch * 32 + T - 1) / T;
  long o = 4L * ND;
  k_gather<<<gB, T, 0, stream>>>(pos_items, items1,    out + o, batch); o += (long)batch * DIMS;
  k_gather<<<gB, T, 0, stream>>>(neg_items, items1,    out + o, batch); o += (long)batch * DIMS;
  k_gather<<<gB, T, 0, stream>>>(pos_items, bitems1,   out + o, batch); o += (long)batch * DIMS;
  k_gather<<<gB, T, 0, stream>>>(neg_items, bitems1,   out + o, batch); o += (long)batch * DIMS;
  k_gather<<<gB, T, 0, stream>>>(users,     users_emb, out + o, batch); o += (long)batch * DIMS;
  k_gather<<<gB, T, 0, stream>>>(baskets,   baskets1,  out + o, batch);
}